// PMField_5196910428903
// MI455X (gfx1250) — compile-verified
//
#include <hip/hip_runtime.h>

typedef __attribute__((ext_vector_type(2))) float v2f;
typedef __attribute__((ext_vector_type(8))) float v8f;

#define EPSC  1e-4f
#define DTB   0.12f     /* DT * BETA */
#define CLMP  3.0f
#define NSTEP 4
#define NC    32
#define ND    8

// One wave handles a 16-point tile.
//   Pass 1 (WMMA f32 16x16x4): dot[c,b] = centers . z   (A = centers, B = z^T)
//   C/D layout: VGPR v, lane l -> (c_local = v + 8*(l>=16), b = l&15)
//   Per-element:  r2 = |z|^2 - 2*dot + |c|^2 + EPS ;  rs = rsqrt(r2)
//                 u = mu*rs (for n), w = mu*rs^3 (for g,s)
//   g[b,d] = sum_c w*cen[c,d] - (sum_c w)*z[b,d] ; reductions over c are
//   in-lane over v + one shfl_xor(16) across wave halves.
__launch_bounds__(256, 4)
__global__ void pmfield_wmma_kernel(const float* __restrict__ z_in,
                                    const float* __restrict__ centers,
                                    const float* __restrict__ mus,
                                    float* __restrict__ z_out)
{
    __shared__ __align__(16) float sc[NC * ND];  // centers, row-major [c][d]
    __shared__ float smu[NC];
    __shared__ float scn[NC];                    // |center|^2 + EPS

    const int tid = threadIdx.x;
    sc[tid & (NC * ND - 1)] = centers[tid & (NC * ND - 1)];   // 256 threads == 256 elems
    if (tid < NC) smu[tid] = mus[tid];
    __syncthreads();
    if (tid < NC) {
        float s = EPSC;
        #pragma unroll
        for (int d = 0; d < ND; ++d) { float c = sc[tid * ND + d]; s = fmaf(c, c, s); }
        scn[tid] = s;
    }
    __syncthreads();

    const int lane = tid & 31;
    const int wv   = tid >> 5;
    const int h    = lane >> 4;    // wave half
    const int nb   = lane & 15;    // point index within tile (N of WMMA)
    const int point = blockIdx.x * 128 + wv * 16 + nb;

    // A-matrix constants (centers): lane m=nb holds A[m, 4*ks+2h] / A[m, 4*ks+2h+1]
    v2f aA[2][2];
    #pragma unroll
    for (int blk = 0; blk < 2; ++blk) {
        #pragma unroll
        for (int ks = 0; ks < 2; ++ks) {
            const int c  = nb + 16 * blk;
            const int d0 = 4 * ks + 2 * h;
            aA[blk][ks].x = sc[c * ND + d0];
            aA[blk][ks].y = sc[c * ND + d0 + 1];
        }
    }

    // per-(v,blk) element constants: this lane's c index is v + 8h + 16blk
    float mu_r[2][8], cn_r[2][8];
    #pragma unroll
    for (int blk = 0; blk < 2; ++blk) {
        #pragma unroll
        for (int v = 0; v < 8; ++v) {
            const int c  = v + 8 * h + 16 * blk;
            mu_r[blk][v] = smu[c];
            cn_r[blk][v] = scn[c];
        }
    }

    // full z row for this lane's point (duplicated across halves)
    const float4* zin4 = reinterpret_cast<const float4*>(z_in);
    float4 zl = zin4[point * 2];
    float4 zh = zin4[point * 2 + 1];
    float z0 = zl.x, z1 = zl.y, z2 = zl.z, z3 = zl.w;
    float z4 = zh.x, z5 = zh.y, z6 = zh.z, z7 = zh.w;

    const float4* sc4 = reinterpret_cast<const float4*>(sc);

    #pragma unroll
    for (int step = 0; step < NSTEP; ++step) {
        float znorm = z0 * z0;
        znorm = fmaf(z1, z1, znorm); znorm = fmaf(z2, z2, znorm);
        znorm = fmaf(z3, z3, znorm); znorm = fmaf(z4, z4, znorm);
        znorm = fmaf(z5, z5, znorm); znorm = fmaf(z6, z6, znorm);
        znorm = fmaf(z7, z7, znorm);

        // B-matrix regs: lane n=nb holds B[4*ks+2h, nb] = z[nb, 4*ks+2h]
        v2f bB[2];
        bB[0].x = h ? z2 : z0;  bB[0].y = h ? z3 : z1;
        bB[1].x = h ? z6 : z4;  bB[1].y = h ? z7 : z5;

        float g0 = 0.f, g1 = 0.f, g2 = 0.f, g3 = 0.f;
        float g4 = 0.f, g5 = 0.f, g6 = 0.f, g7 = 0.f;
        float sp = 0.f, nnp = 0.f;

        #pragma unroll
        for (int blk = 0; blk < 2; ++blk) {
            v8f acc = {0.f, 0.f, 0.f, 0.f, 0.f, 0.f, 0.f, 0.f};
            acc = __builtin_amdgcn_wmma_f32_16x16x4_f32(
                    false, aA[blk][0], false, bB[0], (short)0, acc, false, false);
            acc = __builtin_amdgcn_wmma_f32_16x16x4_f32(
                    false, aA[blk][1], false, bB[1], (short)0, acc, false, false);

            #pragma unroll
            for (int v = 0; v < 8; ++v) {
                const float dot = acc[v];
                float r2 = fmaf(-2.f, dot, znorm) + cn_r[blk][v];
                float rs = __builtin_amdgcn_rsqf(r2);     // 1/r
                float u  = mu_r[blk][v] * rs;             // mu/r
                float w  = u * rs * rs;                   // mu/(r*r2)
                nnp += u;
                sp  += w;
                const int c = v + 8 * h + 16 * blk;
                const float4 c0 = sc4[c * 2];
                const float4 c1 = sc4[c * 2 + 1];
                g0 = fmaf(w, c0.x, g0); g1 = fmaf(w, c0.y, g1);
                g2 = fmaf(w, c0.z, g2); g3 = fmaf(w, c0.w, g3);
                g4 = fmaf(w, c1.x, g4); g5 = fmaf(w, c1.y, g5);
                g6 = fmaf(w, c1.z, g6); g7 = fmaf(w, c1.w, g7);
            }
        }

        // combine the two wave halves (each covered half of the centers)
        nnp += __shfl_xor(nnp, 16, 32);
        sp  += __shfl_xor(sp,  16, 32);
        g0  += __shfl_xor(g0,  16, 32);
        g1  += __shfl_xor(g1,  16, 32);
        g2  += __shfl_xor(g2,  16, 32);
        g3  += __shfl_xor(g3,  16, 32);
        g4  += __shfl_xor(g4,  16, 32);
        g5  += __shfl_xor(g5,  16, 32);
        g6  += __shfl_xor(g6,  16, 32);
        g7  += __shfl_xor(g7,  16, 32);

        const float nn    = 1.0f + nnp;
        const float scale = DTB * __builtin_amdgcn_rcpf(nn);

        // z = clamp(z + scale * (g - sp*z), -3, 3)
        float t;
        t = fmaf(-sp, z0, g0); z0 = fmaf(scale, t, z0); z0 = fminf(CLMP, fmaxf(-CLMP, z0));
        t = fmaf(-sp, z1, g1); z1 = fmaf(scale, t, z1); z1 = fminf(CLMP, fmaxf(-CLMP, z1));
        t = fmaf(-sp, z2, g2); z2 = fmaf(scale, t, z2); z2 = fminf(CLMP, fmaxf(-CLMP, z2));
        t = fmaf(-sp, z3, g3); z3 = fmaf(scale, t, z3); z3 = fminf(CLMP, fmaxf(-CLMP, z3));
        t = fmaf(-sp, z4, g4); z4 = fmaf(scale, t, z4); z4 = fminf(CLMP, fmaxf(-CLMP, z4));
        t = fmaf(-sp, z5, g5); z5 = fmaf(scale, t, z5); z5 = fminf(CLMP, fmaxf(-CLMP, z5));
        t = fmaf(-sp, z6, g6); z6 = fmaf(scale, t, z6); z6 = fminf(CLMP, fmaxf(-CLMP, z6));
        t = fmaf(-sp, z7, g7); z7 = fmaf(scale, t, z7); z7 = fminf(CLMP, fmaxf(-CLMP, z7));
    }

    if (h == 0) {   // halves are duplicates; write once
        float4* out4 = reinterpret_cast<float4*>(z_out);
        out4[point * 2]     = make_float4(z0, z1, z2, z3);
        out4[point * 2 + 1] = make_float4(z4, z5, z6, z7);
    }
}

extern "C" void kernel_launch(void* const* d_in, const int* in_sizes, int n_in,
                              void* d_out, int out_size, void* d_ws, size_t ws_size,
                              hipStream_t stream) {
    (void)n_in; (void)out_size; (void)d_ws; (void)ws_size;
    const float* z       = (const float*)d_in[0];  // (B, 8)
    const float* centers = (const float*)d_in[1];  // (32, 8)
    const float* mus     = (const float*)d_in[2];  // (32,)
    float* out = (float*)d_out;

    const int nPoints = in_sizes[0] / ND;          // B = 524288
    const int blocks  = nPoints / 128;             // 128 points per 256-thread block
    pmfield_wmma_kernel<<<blocks, 256, 0, stream>>>(z, centers, mus, out);
}